// GCN_76888504533336
// MI455X (gfx1250) — compile-verified
//
#include <hip/hip_runtime.h>

typedef __attribute__((ext_vector_type(2))) float v2f;
typedef __attribute__((ext_vector_type(8))) float v8f;

#define FIN    256
#define FOUT   16
#define BLK_ROWS 128      // 8 waves/block, 16 rows per wave
#define KCHUNK 64
#define XPAD   65         // padded LDS row stride (floats) -> conflict-free A reads

// ---------------------------------------------------------------------------
// Degree kernels
// ---------------------------------------------------------------------------
__global__ void k_init_deg(float* dg, int N) {
    int i = blockIdx.x * blockDim.x + threadIdx.x;
    if (i < N) dg[i] = 1.0f;                      // self-loop contributes 1
}

__global__ void k_deg_accum(const int* __restrict__ dst, float* dg, int E) {
    int e = blockIdx.x * blockDim.x + threadIdx.x;
    if (e < E) atomicAdd(&dg[dst[e]], 1.0f);
}

__global__ void k_deg_rsqrt(float* dg, int N) {
    int i = blockIdx.x * blockDim.x + threadIdx.x;
    if (i < N) dg[i] = rsqrtf(dg[i]);             // deg >= 1 always (self-loop)
}

// ---------------------------------------------------------------------------
// GEMM1: h[N,16] = x[N,256] @ W1[256,16]  via V_WMMA_F32_16X16X4_F32
// ---------------------------------------------------------------------------
__global__ void __launch_bounds__(256) k_gemm_wmma(const float* __restrict__ x,
                                                   const float* __restrict__ W1,
                                                   float* __restrict__ h, int N) {
    __shared__ float sX[BLK_ROWS * XPAD];   // 33,280 B
    __shared__ float sW[KCHUNK * FOUT];     //  4,096 B

    const int tid  = threadIdx.x;
    const int lane = tid & 31;
    const int wave = tid >> 5;
    const int half = lane >> 4;             // 0: lanes 0-15, 1: lanes 16-31
    const int l16  = lane & 15;
    const int rowBase0 = blockIdx.x * BLK_ROWS;
    const int waveRow  = rowBase0 + wave * 16;

    v8f acc = {};

    for (int c = 0; c < FIN / KCHUNK; ++c) {
        const int k0 = c * KCHUNK;

        // Stage 128x64 chunk of x (coalesced float4), zero-fill OOB rows.
        for (int idx = tid; idx < BLK_ROWS * (KCHUNK / 4); idx += 256) {
            int r  = idx / (KCHUNK / 4);
            int kq = idx % (KCHUNK / 4);
            int gr = rowBase0 + r;
            float4 v = make_float4(0.f, 0.f, 0.f, 0.f);
            if (gr < N) v = *(const float4*)(x + (size_t)gr * FIN + k0 + kq * 4);
            float* p = &sX[r * XPAD + kq * 4];
            p[0] = v.x; p[1] = v.y; p[2] = v.z; p[3] = v.w;
        }
        // Stage 64x16 chunk of W1 (exactly 1 float4 per thread).
        for (int idx = tid; idx < KCHUNK * FOUT / 4; idx += 256) {
            *(float4*)(&sW[idx * 4]) = *(const float4*)(W1 + k0 * FOUT + idx * 4);
        }
        __syncthreads();

        if (waveRow < N) {   // wave-uniform: EXEC stays all-1 for WMMA
            #pragma unroll
            for (int kk = 0; kk < KCHUNK; kk += 4) {
                const int ka = kk + half * 2;
                v2f a, b;
                const float* ap = &sX[(wave * 16 + l16) * XPAD + ka];
                a.x = ap[0];
                a.y = ap[1];
                b.x = sW[(ka + 0) * FOUT + l16];
                b.y = sW[(ka + 1) * FOUT + l16];
                acc = __builtin_amdgcn_wmma_f32_16x16x4_f32(
                        false, a, false, b, (short)0, acc, false, false);
            }
        }
        __syncthreads();
    }

    if (waveRow < N) {
        #pragma unroll
        for (int v = 0; v < 8; ++v) {
            int m = v + half * 8;
            h[(size_t)(waveRow + m) * FOUT + l16] = acc[v];
        }
    }
}

// ---------------------------------------------------------------------------
// Layer 1 aggregation:  agg1 = b1 + selfloop + sum_edges  -> (relu later)
// ---------------------------------------------------------------------------
__global__ void k_init_agg1(const float* __restrict__ h, const float* __restrict__ dg,
                            const float* __restrict__ b1, float* agg1, int N) {
    int idx = blockIdx.x * blockDim.x + threadIdx.x;
    if (idx < N * FOUT) {
        int i = idx >> 4, f = idx & 15;
        float nrm = dg[i] * dg[i];
        agg1[idx] = b1[f] + h[idx] * nrm;   // bias + self-loop term
    }
}

__global__ void k_edge_agg1(const int* __restrict__ src, const int* __restrict__ dst,
                            const float* __restrict__ dg, const float* __restrict__ h,
                            float* agg1, int E) {
    int e = blockIdx.x * blockDim.x + threadIdx.x;
    if (e >= E) return;
    int s = src[e], d = dst[e];
    float nrm = dg[s] * dg[d];
    const float4* hs = (const float4*)(h + (size_t)s * FOUT);
    float* ad = agg1 + (size_t)d * FOUT;
    #pragma unroll
    for (int q = 0; q < 4; ++q) {
        float4 v = hs[q];
        atomicAdd(ad + q * 4 + 0, v.x * nrm);
        atomicAdd(ad + q * 4 + 1, v.y * nrm);
        atomicAdd(ad + q * 4 + 2, v.z * nrm);
        atomicAdd(ad + q * 4 + 3, v.w * nrm);
    }
}

// ---------------------------------------------------------------------------
// z[i] = relu(agg1[i]) . W2   (GEMM2 projection fused as dot product)
// ---------------------------------------------------------------------------
__global__ void k_compute_z(const float* __restrict__ agg1, const float* __restrict__ W2,
                            float* z, int N) {
    int i = blockIdx.x * blockDim.x + threadIdx.x;
    if (i >= N) return;
    const float4* av = (const float4*)(agg1 + (size_t)i * FOUT);
    float acc = 0.f;
    #pragma unroll
    for (int q = 0; q < 4; ++q) {
        float4 v = av[q];
        acc += fmaxf(v.x, 0.f) * W2[q * 4 + 0];
        acc += fmaxf(v.y, 0.f) * W2[q * 4 + 1];
        acc += fmaxf(v.z, 0.f) * W2[q * 4 + 2];
        acc += fmaxf(v.w, 0.f) * W2[q * 4 + 3];
    }
    z[i] = acc;
}

// ---------------------------------------------------------------------------
// Layer 2 aggregation into d_out
// ---------------------------------------------------------------------------
__global__ void k_init_out(const float* __restrict__ z, const float* __restrict__ dg,
                           const float* __restrict__ b2, float* out, int N) {
    int i = blockIdx.x * blockDim.x + threadIdx.x;
    if (i < N) out[i] = b2[0] + z[i] * dg[i] * dg[i];  // bias + self-loop
}

__global__ void k_edge_agg2(const int* __restrict__ src, const int* __restrict__ dst,
                            const float* __restrict__ dg, const float* __restrict__ z,
                            float* out, int E) {
    int e = blockIdx.x * blockDim.x + threadIdx.x;
    if (e >= E) return;
    int s = src[e], d = dst[e];
    atomicAdd(&out[d], z[s] * dg[s] * dg[d]);
}

// ---------------------------------------------------------------------------
extern "C" void kernel_launch(void* const* d_in, const int* in_sizes, int n_in,
                              void* d_out, int out_size, void* d_ws, size_t ws_size,
                              hipStream_t stream) {
    const float* x  = (const float*)d_in[0];
    const int*   ei = (const int*)d_in[1];
    const float* W1 = (const float*)d_in[2];
    const float* b1 = (const float*)d_in[3];
    const float* W2 = (const float*)d_in[4];
    const float* b2 = (const float*)d_in[5];
    float* out = (float*)d_out;

    const int N = in_sizes[0] / FIN;     // 100000
    const int E = in_sizes[1] / 2;       // 3200000
    const int* src = ei;
    const int* dst = ei + E;

    // workspace layout (floats): dg[N] | h[N*16] | agg1[N*16] | z[N]  = 34N floats
    float* ws   = (float*)d_ws;
    float* dg   = ws;
    float* h    = ws + (size_t)N;
    float* agg1 = ws + (size_t)N * 17;
    float* z    = ws + (size_t)N * 33;

    const int T = 256;
    // degrees
    k_init_deg <<<(N + T - 1) / T, T, 0, stream>>>(dg, N);
    k_deg_accum<<<(E + T - 1) / T, T, 0, stream>>>(dst, dg, E);
    k_deg_rsqrt<<<(N + T - 1) / T, T, 0, stream>>>(dg, N);
    // GEMM1 via fp32 WMMA
    k_gemm_wmma<<<(N + BLK_ROWS - 1) / BLK_ROWS, T, 0, stream>>>(x, W1, h, N);
    // layer-1 aggregation
    k_init_agg1<<<((size_t)N * FOUT + T - 1) / T, T, 0, stream>>>(h, dg, b1, agg1, N);
    k_edge_agg1<<<(E + T - 1) / T, T, 0, stream>>>(src, dst, dg, h, agg1, E);
    // relu + W2 projection
    k_compute_z<<<(N + T - 1) / T, T, 0, stream>>>(agg1, W2, z, N);
    // layer-2 aggregation
    k_init_out <<<(N + T - 1) / T, T, 0, stream>>>(z, dg, b2, out, N);
    k_edge_agg2<<<(E + T - 1) / T, T, 0, stream>>>(src, dst, dg, z, out, E);
}